// GNN_22308060136020
// MI455X (gfx1250) — compile-verified
//
#include <hip/hip_runtime.h>
#include <math.h>

#define NEG_SLOPE 0.01f

typedef float v2f __attribute__((ext_vector_type(2)));
typedef float v8f __attribute__((ext_vector_type(8)));

__device__ __forceinline__ float leaky(float v) { return v >= 0.f ? v : NEG_SLOPE * v; }

// ---------------------------------------------------------------------------
// Utility kernels
// ---------------------------------------------------------------------------
__global__ void zero_kernel(float* p, long long n) {
    long long i = (long long)blockIdx.x * blockDim.x + threadIdx.x;
    long long stride = (long long)gridDim.x * blockDim.x;
    for (; i < n; i += stride) p[i] = 0.f;
}

__global__ void copy_kernel(float* __restrict__ dst, const float* __restrict__ src, long long n) {
    long long i = (long long)blockIdx.x * blockDim.x + threadIdx.x;
    long long stride = (long long)gridDim.x * blockDim.x;
    for (; i < n; i += stride) dst[i] = src[i];
}

// ---------------------------------------------------------------------------
// fp32 WMMA GEMM: C[M, NT*16] = A[M,K] @ B[K, NT*16]  (row-major, M%16==0, K%32==0)
//
// Block = 256 threads = 8 waves. Each wave owns one 16-row M-tile and ALL NT
// column tiles (NT v8f accumulators). B is staged chunk-wise (KC x N) in LDS
// cooperatively by the block, so B fragments come from ds_load shared by all
// 8 waves, and each A fragment (one global b64 per 4 K-steps) is reused across
// all NT WMMAs. act: 0 = none, 1 = tanh(v+bias), 2 = leaky(v+bias).
// ---------------------------------------------------------------------------
template <int NT>
__global__ void wmma_gemm_f32(const float* __restrict__ A, int lda,
                              const float* __restrict__ B, int ldb,
                              float* __restrict__ C, int ldc,
                              int M, int K,
                              const float* __restrict__ bias, int act) {
    constexpr int KC = 32;
    constexpr int N  = NT * 16;
    __shared__ float Bs[KC * N];

    int wave   = threadIdx.x >> 5;
    int mtile  = blockIdx.x * 8 + wave;
    int tilesM = M >> 4;
    bool valid = (mtile < tilesM);            // wave-uniform: EXEC all-1s inside
    int lane = threadIdx.x & 31;
    int half = lane >> 4;                     // 0: K=k..k+1, 1: K=k+2..k+3
    int l16  = lane & 15;

    const float* Arow = A + (size_t)((valid ? mtile : 0) * 16 + l16) * lda;

    v8f acc[NT];
#pragma unroll
    for (int t = 0; t < NT; ++t) acc[t] = (v8f){0.f, 0.f, 0.f, 0.f, 0.f, 0.f, 0.f, 0.f};

    for (int k0 = 0; k0 < K; k0 += KC) {
        // cooperative stage of B[k0 .. k0+KC) x N into LDS (coalesced)
        for (int i = threadIdx.x; i < KC * N; i += blockDim.x) {
            int r = i / N, c = i - r * N;
            Bs[i] = B[(size_t)(k0 + r) * ldb + c];
        }
        __syncthreads();
        if (valid) {
            for (int kk = 0; kk < KC; kk += 4) {
                int ka = kk + 2 * half;
                v2f a;                        // A 16x4 fragment: row l16, K = ka, ka+1
                a.x = Arow[k0 + ka];
                a.y = Arow[k0 + ka + 1];
#pragma unroll
                for (int t = 0; t < NT; ++t) {
                    v2f b;                    // B 4x16 fragment from LDS
                    b.x = Bs[ka * N + t * 16 + l16];
                    b.y = Bs[(ka + 1) * N + t * 16 + l16];
                    acc[t] = __builtin_amdgcn_wmma_f32_16x16x4_f32(
                        false, a, false, b, (short)0, acc[t], false, false);
                }
            }
        }
        __syncthreads();
    }

    if (valid) {
#pragma unroll
        for (int t = 0; t < NT; ++t) {
            int n = t * 16 + l16;
            float bb = bias ? bias[n] : 0.f;
#pragma unroll
            for (int r = 0; r < 8; ++r) {     // C: VGPR r -> rows r and r+8
                int m = mtile * 16 + r + half * 8;
                float v = acc[t][r];
                if (act == 1)      v = tanhf(v + bb);
                else if (act == 2) v = leaky(v + bb);
                C[(size_t)m * ldc + n] = v;
            }
        }
    }
}

// ---------------------------------------------------------------------------
// Row-wise L2 normalize (in place), one wave per row; optional leaky_relu.
// ---------------------------------------------------------------------------
__global__ void row_l2norm(float* X, int ld, int M, int D, int applyLeaky) {
    int row = blockIdx.x * (blockDim.x >> 5) + (threadIdx.x >> 5);
    if (row >= M) return;
    int lane = threadIdx.x & 31;
    float* p = X + (size_t)row * ld;
    float ss = 0.f;
    for (int c = lane; c < D; c += 32) { float v = p[c]; ss += v * v; }
    for (int o = 16; o; o >>= 1) ss += __shfl_down(ss, o, 32);
    ss = __shfl(ss, 0, 32);
    float inv = 1.f / fmaxf(sqrtf(ss), 1e-12f);
    for (int c = lane; c < D; c += 32) {
        float v = p[c] * inv;
        if (applyLeaky) v = leaky(v);
        p[c] = v;
    }
}

// ---------------------------------------------------------------------------
// Graph attention edge kernels. edge_index is [2, E] int32 (JAX default x32).
// ---------------------------------------------------------------------------
__global__ void deg_kernel(const int* __restrict__ ei, int E, float* deg) {
    int e = blockIdx.x * blockDim.x + threadIdx.x;
    if (e < E) atomicAdd(&deg[ei[e]], 1.f);       // src row
}

// encoded-uint ordering: monotone map float -> uint so atomicMax works
__device__ __forceinline__ unsigned enc_f32(float f) {
    unsigned b = __float_as_uint(f);
    return (b & 0x80000000u) ? ~b : (b | 0x80000000u);
}
__device__ __forceinline__ float dec_f32(unsigned e) {
    unsigned b = (e & 0x80000000u) ? (e & 0x7FFFFFFFu) : ~e;
    return __uint_as_float(b);
}

// one wave per edge: inner = <xw[dst], xw[src]>, gate = sigmoid(deg(src)^-.5 * inner)
__global__ void edge_score(const float* __restrict__ xw, int D,
                           const int* __restrict__ ei, int E,
                           const float* __restrict__ deg,
                           float* __restrict__ score,
                           unsigned* __restrict__ smax) {
    int e = blockIdx.x * (blockDim.x >> 5) + (threadIdx.x >> 5);
    if (e >= E) return;
    int lane = threadIdx.x & 31;
    int s = ei[e];
    int d = ei[E + e];
    const float* xs = xw + (size_t)s * D;
    const float* xd = xw + (size_t)d * D;
    float sum = 0.f;
    for (int c = lane; c < D; c += 32) sum += xd[c] * xs[c];
    for (int o = 16; o; o >>= 1) sum += __shfl_down(sum, o, 32);
    sum = __shfl(sum, 0, 32);
    if (lane == 0) {
        float g  = 1.f / (1.f + expf(-rsqrtf(deg[s]) * sum));
        float sc = sum * g;
        score[e] = sc;
        atomicMax(&smax[d], enc_f32(sc));
    }
}

__global__ void edge_denom(const int* __restrict__ ei, int E,
                           float* __restrict__ score,
                           const unsigned* __restrict__ smax,
                           float* __restrict__ denom) {
    int e = blockIdx.x * blockDim.x + threadIdx.x;
    if (e >= E) return;
    int d = ei[E + e];
    float w = expf(score[e] - dec_f32(smax[d]));
    score[e] = w;                                  // reuse as exp-weight
    atomicAdd(&denom[d], w);
}

// one wave per edge: out[dst] += xw[src] * alpha
__global__ void edge_aggregate(const float* __restrict__ xw, int D,
                               const int* __restrict__ ei, int E,
                               const float* __restrict__ score,
                               const float* __restrict__ denom,
                               float* __restrict__ outAgg) {
    int e = blockIdx.x * (blockDim.x >> 5) + (threadIdx.x >> 5);
    if (e >= E) return;
    int lane = threadIdx.x & 31;
    int s = ei[e];
    int d = ei[E + e];
    float w = score[e] / (denom[d] + 1e-16f);
    const float* xs = xw + (size_t)s * D;
    float* od = outAgg + (size_t)d * D;
    for (int c = lane; c < D; c += 32) atomicAdd(&od[c], xs[c] * w);
}

// ---------------------------------------------------------------------------
// x_next = leaky(h@Gw + gb + leaky(x@Lw + lb) + id_emb); also write into
// d_out at column offset (output is concat([x1, x2], axis=1), ld = 128).
// ---------------------------------------------------------------------------
__global__ void combine_kernel(const float* __restrict__ hg,
                               const float* __restrict__ xl,
                               const float* __restrict__ gb,
                               const float* __restrict__ lb,
                               const float* __restrict__ id_emb,
                               float* __restrict__ x_next,   // may be null
                               float* __restrict__ out, int colOff, int Nn) {
    int i = blockIdx.x * blockDim.x + threadIdx.x;
    if (i >= Nn * 64) return;
    int n = i >> 6, c = i & 63;
    float xh = leaky(xl[i] + lb[c]) + id_emb[i];
    float v  = leaky(hg[i] + gb[c] + xh);
    if (x_next) x_next[i] = v;
    out[(size_t)n * 128 + colOff + c] = v;
}

// ---------------------------------------------------------------------------
// Host-side pipeline
// ---------------------------------------------------------------------------
extern "C" void kernel_launch(void* const* d_in, const int* in_sizes, int n_in,
                              void* d_out, int out_size, void* d_ws, size_t ws_size,
                              hipStream_t stream) {
    const int NU = 50000, NI = 30000, NN = NU + NI, E = 1000000;
    (void)in_sizes; (void)n_in; (void)out_size; (void)ws_size;

    const float* features = (const float*)d_in[0];   // [30000, 768]
    const float* id_emb   = (const float*)d_in[1];   // [80000, 64]
    const int*   ei       = (const int*)d_in[2];     // [2, 1M] (JAX x32 int)
    const float* pref     = (const float*)d_in[3];   // [50000, 128]
    const float* mlp_w    = (const float*)d_in[4];
    const float* mlp_b    = (const float*)d_in[5];
    const float* conv1_w  = (const float*)d_in[6];
    const float* lin1_w   = (const float*)d_in[7];
    const float* lin1_b   = (const float*)d_in[8];
    const float* g1_w     = (const float*)d_in[9];
    const float* g1_b     = (const float*)d_in[10];
    const float* conv2_w  = (const float*)d_in[11];
    const float* lin2_w   = (const float*)d_in[12];
    const float* lin2_b   = (const float*)d_in[13];
    const float* g2_w     = (const float*)d_in[14];
    const float* g2_b     = (const float*)d_in[15];
    float* out = (float*)d_out;

    float* ws = (float*)d_ws;
    size_t off = 0;
    float*    bufX  = ws + off; off += (size_t)NN * 128;  // x
    float*    bufT  = ws + off; off += (size_t)NN * 128;  // xw / two [NN,64] temps
    float*    bufA  = ws + off; off += (size_t)NN * 128;  // aggregation / h
    float*    bufX1 = ws + off; off += (size_t)NN * 64;   // x1
    float*    score = ws + off; off += (size_t)E;
    float*    deg   = ws + off; off += (size_t)NN;
    unsigned* smax  = (unsigned*)(ws + off); off += (size_t)NN;
    float*    denom = ws + off; off += (size_t)NN;

    const int TPB = 256;
    const int WPB = TPB / 32;
    auto gemmBlocks = [](int M) { return ((M / 16) + 7) / 8; };  // 8 M-tiles per block
    int edgeWaveBlocks = (E + WPB - 1) / WPB;     // wave-per-edge kernels
    int edgeThrBlocks  = (E + TPB - 1) / TPB;
    int rowBlocks      = (NN + WPB - 1) / WPB;

    // ---- degrees (edge-structure only: computed once, reused by both layers)
    zero_kernel<<<1024, TPB, 0, stream>>>(deg, NN);
    deg_kernel<<<edgeThrBlocks, TPB, 0, stream>>>(ei, E, deg);

    // ---- x = l2norm(concat(preference, tanh(features@mlp_w + mlp_b)))
    wmma_gemm_f32<8><<<gemmBlocks(NI), TPB, 0, stream>>>(
        features, 768, mlp_w, 128, bufX + (size_t)NU * 128, 128, NI, 768, mlp_b, 1);
    copy_kernel<<<2048, TPB, 0, stream>>>(bufX, pref, (long long)NU * 128);
    row_l2norm<<<rowBlocks, TPB, 0, stream>>>(bufX, 128, NN, 128, 0);

    // ---- GAT layer 1 (D = 128) ------------------------------------------
    wmma_gemm_f32<8><<<gemmBlocks(NN), TPB, 0, stream>>>(
        bufX, 128, conv1_w, 128, bufT, 128, NN, 128, nullptr, 0);
    zero_kernel<<<4096, TPB, 0, stream>>>(bufA, (long long)NN * 128);
    zero_kernel<<<512, TPB, 0, stream>>>((float*)smax, NN);
    zero_kernel<<<512, TPB, 0, stream>>>(denom, NN);
    edge_score<<<edgeWaveBlocks, TPB, 0, stream>>>(bufT, 128, ei, E, deg, score, smax);
    edge_denom<<<edgeThrBlocks, TPB, 0, stream>>>(ei, E, score, smax, denom);
    edge_aggregate<<<edgeWaveBlocks, TPB, 0, stream>>>(bufT, 128, ei, E, score, denom, bufA);
    row_l2norm<<<rowBlocks, TPB, 0, stream>>>(bufA, 128, NN, 128, 1);  // h = leaky(l2norm(agg))

    // ---- x1 = leaky(h@g1_w + g1_b + leaky(x@lin1_w + lin1_b) + id_emb) ---
    wmma_gemm_f32<4><<<gemmBlocks(NN), TPB, 0, stream>>>(
        bufA, 128, g1_w, 64, bufT, 64, NN, 128, nullptr, 0);
    wmma_gemm_f32<4><<<gemmBlocks(NN), TPB, 0, stream>>>(
        bufX, 128, lin1_w, 64, bufT + (size_t)NN * 64, 64, NN, 128, nullptr, 0);
    combine_kernel<<<(NN * 64 + TPB - 1) / TPB, TPB, 0, stream>>>(
        bufT, bufT + (size_t)NN * 64, g1_b, lin1_b, id_emb, bufX1, out, 0, NN);

    // ---- GAT layer 2 (D = 64) -------------------------------------------
    wmma_gemm_f32<4><<<gemmBlocks(NN), TPB, 0, stream>>>(
        bufX1, 64, conv2_w, 64, bufT, 64, NN, 64, nullptr, 0);
    zero_kernel<<<2048, TPB, 0, stream>>>(bufA, (long long)NN * 64);
    zero_kernel<<<512, TPB, 0, stream>>>((float*)smax, NN);
    zero_kernel<<<512, TPB, 0, stream>>>(denom, NN);
    edge_score<<<edgeWaveBlocks, TPB, 0, stream>>>(bufT, 64, ei, E, deg, score, smax);
    edge_denom<<<edgeThrBlocks, TPB, 0, stream>>>(ei, E, score, smax, denom);
    edge_aggregate<<<edgeWaveBlocks, TPB, 0, stream>>>(bufT, 64, ei, E, score, denom, bufA);
    row_l2norm<<<rowBlocks, TPB, 0, stream>>>(bufA, 64, NN, 64, 1);

    // ---- x2 = leaky(h@g2_w + g2_b + leaky(x1@lin2_w + lin2_b) + id_emb) --
    wmma_gemm_f32<4><<<gemmBlocks(NN), TPB, 0, stream>>>(
        bufA, 64, g2_w, 64, bufT, 64, NN, 64, nullptr, 0);
    wmma_gemm_f32<4><<<gemmBlocks(NN), TPB, 0, stream>>>(
        bufX1, 64, lin2_w, 64, bufT + (size_t)NN * 64, 64, NN, 64, nullptr, 0);
    combine_kernel<<<(NN * 64 + TPB - 1) / TPB, TPB, 0, stream>>>(
        bufT, bufT + (size_t)NN * 64, g2_b, lin2_b, id_emb, nullptr, out, 64, NN);
}